// DetectionConfidenceMap2keypoint_2_43602507989177
// MI455X (gfx1250) — compile-verified
//
#include <hip/hip_runtime.h>
#include <hip/hip_bf16.h>

// ---------------------------------------------------------------------------
// DetectionConfidenceMap2keypoint for MI455X (gfx1250, wave32).
//
// Bandwidth-bound: ~492 MB of mandatory HBM traffic (~21 us @ 23.3 TB/s).
// Pass 1 fuses sigmoid + store + the three per-(b,k) weighted reductions so
// each map is read exactly once and written exactly once. Cross-lane
// reduction of partials uses V_WMMA_F32_16X16X4_F32 (f32 matrix pipe,
// full precision): one WMMA + 8 adds + 1 shfl replaces a 5-step shuffle tree.
// Pass 2 is a tiny 1600-thread gather/keypoint epilogue (L2-resident).
// ---------------------------------------------------------------------------

typedef __attribute__((ext_vector_type(2))) float v2f;
typedef __attribute__((ext_vector_type(8))) float v8f;

#define BB 16
#define KK 100
#define HH 120
#define WW 160
#define HW (HH * WW)          // 19200
#define MAPS (BB * KK)        // 1600
#define MAP_ELEMS ((size_t)MAPS * HW)  // 30,720,000

// Full wave32 reduction through the matrix unit.
// A-matrix 16x4 f32 layout: VGPR0 = {lanes 0-15 -> A[m,0], lanes 16-31 -> A[m,2]},
// VGPR1 = {A[m,1], A[m,3]}. With a = {p, 0} and B = ones:
//   D[m,n] = p[m] + p[m+16]  (independent of n).
// D layout: lane L<16 VGPR j = D[j, L]; lane L>=16 VGPR j = D[8+j, L-16].
// Sum of a lane's 8 D regs = rows 0-7 total (lanes 0-15) or rows 8-15 total
// (lanes 16-31); one xor-16 shuffle-add yields the full 32-lane sum in every lane.
__device__ __forceinline__ float wave_reduce_f32_wmma(float p) {
    v2f a; a.x = p;    a.y = 0.0f;
    v2f b; b.x = 1.0f; b.y = 1.0f;
    v8f c = {};
    v8f d = __builtin_amdgcn_wmma_f32_16x16x4_f32(
        /*neg_a=*/false, a, /*neg_b=*/false, b,
        /*c_mod=*/(short)0, c, /*reuse_a=*/false, /*reuse_b=*/false);
    float s = d[0] + d[1] + d[2] + d[3] + d[4] + d[5] + d[6] + d[7];
    s += __shfl_xor(s, 16, 32);
    return s;
}

__device__ __forceinline__ float sigmoidf(float x) {
    return 1.0f / (1.0f + expf(-x));
}

// One block per (b,k) map. 256 threads = 8 wave32.
// Streams float4: sigmoid -> out, accumulate {sum, sum*x, sum*y}.
__global__ __launch_bounds__(256)
void sigmoid_reduce_kernel(const float* __restrict__ in,
                           float* __restrict__ out,
                           float* __restrict__ red /* [MAPS*3] */) {
    const int map = blockIdx.x;
    const size_t base = (size_t)map * HW;
    const float4* __restrict__ in4  = (const float4*)(in + base);
    float4* __restrict__ out4 = (float4*)(out + base);

    float z = 0.0f, px = 0.0f, py = 0.0f;
    const int n4 = HW / 4;  // 4800; WW % 4 == 0 so a float4 never crosses a row
    for (int u = threadIdx.x; u < n4; u += 256) {
        float4 r = in4[u];
        float4 v;
        v.x = sigmoidf(r.x);
        v.y = sigmoidf(r.y);
        v.z = sigmoidf(r.z);
        v.w = sigmoidf(r.w);
        out4[u] = v;
        const int e = u << 2;
        const int y = e / WW;
        const int x = e - y * WW;
        const float sv = (v.x + v.y) + (v.z + v.w);
        z  += sv;
        px += v.x * (float)x + v.y * (float)(x + 1) +
              v.z * (float)(x + 2) + v.w * (float)(x + 3);
        py += sv * (float)y;
    }

    // Per-wave reduction on the matrix pipe (EXEC all-ones here: no divergence).
    z  = wave_reduce_f32_wmma(z);
    px = wave_reduce_f32_wmma(px);
    py = wave_reduce_f32_wmma(py);

    __shared__ float lz[8], lx[8], ly[8];
    const int wave = threadIdx.x >> 5;
    const int lane = threadIdx.x & 31;
    if (lane == 0) { lz[wave] = z; lx[wave] = px; ly[wave] = py; }
    __syncthreads();
    if (threadIdx.x == 0) {
        float tz = 0.0f, tx = 0.0f, ty = 0.0f;
        #pragma unroll
        for (int w = 0; w < 8; ++w) { tz += lz[w]; tx += lx[w]; ty += ly[w]; }
        red[map * 3 + 0] = tz;
        red[map * 3 + 1] = tx;
        red[map * 3 + 2] = ty;
    }
}

__device__ __forceinline__ int clampi(int v, int lo, int hi) {
    return v < lo ? lo : (v > hi ? hi : v);
}

// One thread per (b,k). score2 map is ALWAYS the Dk branch (bug-faithful).
__global__ __launch_bounds__(256)
void keypoint_kernel(const float* __restrict__ Dmap,     // score/gather map for this branch
                     const float* __restrict__ score2map,// Dk map (both branches)
                     const float* __restrict__ red,      // [MAPS*3]
                     float* __restrict__ kp_out,         // [B, 3K, 3]
                     float* __restrict__ zeta_out) {     // [B, K] or nullptr
    const int idx = blockIdx.x * blockDim.x + threadIdx.x;
    if (idx >= MAPS) return;

    const float zeta = red[idx * 3 + 0];
    const float npx  = red[idx * 3 + 1];
    const float npy  = red[idx * 3 + 2];

    const float kx = rintf(npx / zeta);   // RNE == jnp.round
    const float ky = rintf(npy / zeta);

    const float* __restrict__ m1 = Dmap + (size_t)idx * HW;
    const float* __restrict__ m2 = score2map + (size_t)idx * HW;

    const int gx = clampi((int)kx, 0, WW - 1);
    const int gy = clampi((int)ky, 0, HH - 1);
    const float s = m1[gy * WW + gx];

    // torch .int() truncation toward zero (values are >= 0 here anyway)
    const int kp1x = (int)(kx + kx * s);
    const int kp1y = (int)(ky + ky * s);
    const int kp2x = (int)(kx - kx * s);
    const int kp2y = (int)(ky - ky * s);

    const int w1 = clampi(kp1x, 0, WW - 1);
    const int h1 = clampi(kp1y, 0, HH - 1);
    const int w2 = clampi(kp2x, 0, WW - 1);
    const int h2 = clampi(kp2y, 0, HH - 1);

    const float s1 = m1[h1 * WW + w1];
    const float s2 = m2[h2 * WW + w2];

    const int b = idx / KK;
    const int k = idx - b * KK;
    float* __restrict__ rb = kp_out + (size_t)b * (3 * KK) * 3;

    float* r0 = rb + (size_t)k * 3;                  // kp_w
    r0[0] = kx;           r0[1] = ky;           r0[2] = s;
    float* r1 = rb + (size_t)(KK + k) * 3;           // kp1_w
    r1[0] = (float)kp1x;  r1[1] = (float)kp1y;  r1[2] = s1;
    float* r2 = rb + (size_t)(2 * KK + k) * 3;       // kp2_w
    r2[0] = (float)kp2x;  r2[1] = (float)kp2y;  r2[2] = s2;

    if (zeta_out) zeta_out[idx] = zeta;
}

extern "C" void kernel_launch(void* const* d_in, const int* in_sizes, int n_in,
                              void* d_out, int out_size, void* d_ws, size_t ws_size,
                              hipStream_t stream) {
    (void)in_sizes; (void)n_in; (void)out_size; (void)ws_size;

    const float* Rk    = (const float*)d_in[0];
    const float* tf_Rk = (const float*)d_in[1];

    float* out      = (float*)d_out;
    float* Dk       = out;                              // 30,720,000
    float* keypoint = Dk + MAP_ELEMS;                   // 16*300*3 = 14,400
    float* zeta     = keypoint + (size_t)BB * 3 * KK * 3; // 1,600
    float* tf_Dk    = zeta + (size_t)BB * KK;           // 30,720,000
    float* tf_kp    = tf_Dk + MAP_ELEMS;                // 14,400

    float* red0 = (float*)d_ws;          // MAPS*3 floats
    float* red1 = red0 + MAPS * 3;       // MAPS*3 floats

    // Pass 1: fused sigmoid + store + weighted reductions (one pass per map).
    sigmoid_reduce_kernel<<<MAPS, 256, 0, stream>>>(Rk, Dk, red0);
    sigmoid_reduce_kernel<<<MAPS, 256, 0, stream>>>(tf_Rk, tf_Dk, red1);

    // Pass 2: keypoint epilogue. score2 is gathered from Dk in BOTH branches.
    const int kblocks = (MAPS + 255) / 256;
    keypoint_kernel<<<kblocks, 256, 0, stream>>>(Dk,    Dk, red0, keypoint, zeta);
    keypoint_kernel<<<kblocks, 256, 0, stream>>>(tf_Dk, Dk, red1, tf_kp,   nullptr);
}